// MambaTimeSeriesClassifier_V2_30064771072280
// MI455X (gfx1250) — compile-verified
//
#include <hip/hip_runtime.h>
#include <hip/hip_bf16.h>
#include <math.h>

// ---------------------------------------------------------------------------
// Mamba time-series classifier forward, CDNA5 (gfx1250), wave32.
// All GEMMs use V_WMMA_F32_16X16X4_F32 (f32 A/B/C) to exactly match the f32
// reference math. Fragment layouts per CDNA5 ISA 7.12.2:
//   A (16x4):  lanes 0-15 -> M=lane, K=kk..kk+1 (kk = (lane>=16)?2:0 in v0/v1)
//   B (4x16):  lanes 0-15 -> N=lane, rows K=kk / kk+1 in v0/v1
//   C/D(16x16): vgpr e, lane -> M = e + 8*(lane>>4), N = lane&15
// All fragment loads are UNCONDITIONAL (clamped addresses / don't-care rows)
// so the inner loops are pure load->wmma streams with no exec masking.
// ---------------------------------------------------------------------------

typedef __attribute__((ext_vector_type(2))) float v2f;
typedef __attribute__((ext_vector_type(8))) float v8f;

__device__ __forceinline__ v8f wmma4(v2f a, v2f b, v8f c) {
  // (neg_a, A, neg_b, B, c_mod, C, reuse_a, reuse_b)
  return __builtin_amdgcn_wmma_f32_16x16x4_f32(false, a, false, b, (short)0, c,
                                               false, false);
}

// Branchless, numerically stable softplus: log1p(e^x) = max(x,0)+log1p(e^-|x|)
// (identical decomposition to jax.nn.softplus / logaddexp(x, 0)).
__device__ __forceinline__ float softplusf_(float x) {
  return fmaxf(x, 0.f) + log1pf(__expf(-fabsf(x)));
}
__device__ __forceinline__ float siluf_(float x) {
  return x / (1.f + __expf(-x));
}

#define Z8 {0.f,0.f,0.f,0.f,0.f,0.f,0.f,0.f}

// ===========================================================================
// Kernel 1: one workgroup per (region r, bt) problem. 256 thr = 8 waves.
//   proj_dt (49x24) and proj_xBC (49x896) via WMMA (K=192),
//   depthwise conv(4)+SiLU, softplus+scan, head-reduced xs, then
//   agg = xs^T(32x64pad) @ B(64pad x128) via WMMA.
// ===========================================================================
__global__ __launch_bounds__(256) void k1_region(
    const float* __restrict__ inputs, const float* __restrict__ W_in_r,
    const float* __restrict__ conv_w_r, const float* __restrict__ conv_b_r,
    const float* __restrict__ dt_bias_r, const float* __restrict__ A_log_r,
    float* __restrict__ agg) {
  __shared__ float w_s[64][32];    // dt -> w weights (cols>=24 / rows>=49 junk, unread)
  __shared__ float xs_s[64][32];   // head-reduced x (pad rows stay zero)
  __shared__ float B_s[64][128];   // B matrix (pad rows stay zero)
  __shared__ float tile_s[64][32]; // current 32-channel proj tile (pad rows junk)
  __shared__ int   rowbase[64];    // global float index of u row (valid for all 64)

  const int tid = threadIdx.x;
  const int lane = tid & 31, wave = tid >> 5;
  const int r = blockIdx.x >> 8, bt = blockIdx.x & 255;
  const int r0 = (r >> 1) * 7, c0 = (r & 1) * 7;

  for (int m = tid; m < 64; m += 256) {
    int mm = (m < 49) ? m : 0;  // pad rows alias row 0: their D rows are never read
    int i = mm / 7, j = mm % 7;
    int token = 1 + (r0 + i) * 14 + (c0 + j);
    rowbase[m] = (bt * 197 + token) * 192;
  }
  for (int i = tid; i < 64 * 32; i += 256) (&xs_s[0][0])[i] = 0.f;
  for (int i = tid; i < 64 * 128; i += 256) (&B_s[0][0])[i] = 0.f;
  __syncthreads();

  const float* Wr = W_in_r + (size_t)r * 192 * 1816;
  const int kk = (lane >> 4) * 2;
  const int ln16 = lane & 15;

  // ---- dt projection: W cols 1792..1815, 8 wave-jobs (4 Mtiles x 2 Ntiles)
  {
    int mt = wave >> 1, nt = wave & 1;
    int m0 = mt * 16, n0 = nt * 16;
    const float* uptr = inputs + rowbase[m0 + ln16] + kk;
    int nloc = n0 + ln16;
    int wcol = 1792 + ((nloc < 24) ? nloc : 23);  // clamp: loads always in-bounds
    const float* wptr = Wr + (size_t)kk * 1816 + wcol;
    v8f c = Z8;
#pragma unroll 4
    for (int k0 = 0; k0 < 192; k0 += 4) {
      v2f a, b;
      float2 av = *(const float2*)(uptr + k0);
      a.x = av.x; a.y = av.y;
      b.x = wptr[(size_t)k0 * 1816];
      b.y = wptr[(size_t)k0 * 1816 + 1816];
      c = wmma4(a, b, c);
    }
#pragma unroll
    for (int e = 0; e < 8; ++e) {
      int row = m0 + e + 8 * (lane >> 4);
      w_s[row][nloc] = c[e];  // junk cols/rows never read
    }
  }
  __syncthreads();

  // ---- softplus + scan (24 heads, 49 steps, serial per head)
  if (tid < 24) {
    int h = tid;
    float Ah = -__expf(A_log_r[r * 24 + h]);
    float db = dt_bias_r[r * 24 + h];
    float Stot = 0.f;
    for (int l = 0; l < 49; ++l) {
      float dt = softplusf_(w_s[l][h] + db);
      w_s[l][h] = dt;
      Stot += dt * Ah;
    }
    float S = 0.f;
    for (int l = 0; l < 49; ++l) {
      float dt = w_s[l][h];
      S += dt * Ah;
      w_s[l][h] = __expf(Stot - S) * dt;
    }
  }
  __syncthreads();

  // ---- channel tiles: 28 tiles x 32 channels (0..895; C part unused)
  const float* cw = conv_w_r + (size_t)r * 1024 * 4;
  const float* cb = conv_b_r + (size_t)r * 1024;
  for (int ct = 0; ct < 28; ++ct) {
    {  // proj tile into tile_s (8 wave-jobs)
      int mt = wave >> 1, nt = wave & 1;
      int m0 = mt * 16;
      int nloc = nt * 16 + ln16;
      const float* uptr = inputs + rowbase[m0 + ln16] + kk;
      const float* wptr = Wr + (size_t)kk * 1816 + 768 + ct * 32 + nloc;
      v8f c = Z8;
#pragma unroll 4
      for (int k0 = 0; k0 < 192; k0 += 4) {
        v2f a, b;
        float2 av = *(const float2*)(uptr + k0);
        a.x = av.x; a.y = av.y;
        b.x = wptr[(size_t)k0 * 1816];
        b.y = wptr[(size_t)k0 * 1816 + 1816];
        c = wmma4(a, b, c);
      }
#pragma unroll
      for (int e = 0; e < 8; ++e) {
        int row = m0 + e + 8 * (lane >> 4);
        tile_s[row][nloc] = c[e];  // rows>=49 junk, never read by conv
      }
    }
    __syncthreads();
    // conv(4) + SiLU, then head-weighted accumulate (x) or store (B)
    for (int t = tid; t < 49 * 32; t += 256) {
      int l = t >> 5, cc = t & 31;
      int cg = ct * 32 + cc;
      float acc = cb[cg];
#pragma unroll
      for (int k = 0; k < 4; ++k) {
        int ls = l - 3 + k;
        int lsc = (ls >= 0) ? ls : 0;
        float msk = (ls >= 0) ? 1.f : 0.f;
        acc += cw[cg * 4 + k] * tile_s[lsc][cc] * msk;
      }
      float val = siluf_(acc);
      if (ct < 24) {
        xs_s[l][cc] += w_s[l][ct] * val;  // head index == ct (32-aligned)
      } else {
        B_s[l][cg - 768] = val;
      }
    }
    __syncthreads();
  }

  // ---- agg(32x128) = xs^T @ B  (K = 64 padded rows), 16 wave-jobs
  float* aggp = agg + (size_t)(r * 256 + bt) * 32 * 128;
  for (int j = wave; j < 16; j += 8) {
    int mt = j >> 3, nt = j & 7;
    int m0 = mt * 16, n0 = nt * 16;
    int ml = m0 + ln16, nl = n0 + ln16;
    v8f c = Z8;
#pragma unroll 4
    for (int k0 = 0; k0 < 64; k0 += 4) {
      v2f a, b;
      a.x = xs_s[k0 + kk][ml];
      a.y = xs_s[k0 + kk + 1][ml];
      b.x = B_s[k0 + kk][nl];
      b.y = B_s[k0 + kk + 1][nl];
      c = wmma4(a, b, c);
    }
#pragma unroll
    for (int e = 0; e < 8; ++e) {
      int row = m0 + e + 8 * (lane >> 4);
      aggp[row * 128 + nl] = c[e];
    }
  }
}

// ===========================================================================
// Kernel 2: stage-2 projection. proj2[b,l,n] (n<640 -> W col 512+n,
// n in [640,656) -> W col 1280+(n-640)).  frame is a permutation of agg.
// Grid (64 Mtiles, 11 Ntiles, 8 batches), 64x64 tile per workgroup.
// ===========================================================================
__global__ __launch_bounds__(256) void k2_proj(
    const float* __restrict__ agg, const float* __restrict__ W_in_s,
    float* __restrict__ proj2) {
  __shared__ float A_s[64][129];  // frame tile, +1 pad for bank-free A reads
  const int tid = threadIdx.x, lane = tid & 31, wave = tid >> 5;
  const int b = blockIdx.z;
  const int m_base = blockIdx.x * 64;
  const int n_base = blockIdx.y * 64;

  for (int i = tid; i < 64 * 128; i += 256) {
    int row = i >> 7, col = i & 127;
    int l = m_base + row;
    int t = l >> 7, rr = (l >> 5) & 3, p = l & 31;
    A_s[row][col] = agg[(((size_t)(rr * 256 + b * 32 + t)) * 32 + p) * 128 + col];
  }
  __syncthreads();

  const int kk = (lane >> 4) * 2;
  const int ln16 = lane & 15;
  for (int j = wave * 2; j < wave * 2 + 2; ++j) {
    int mt = j >> 2, nt = j & 3;
    int m0 = mt * 16, n0 = nt * 16;
    int ml = m0 + ln16;
    int gn = n_base + n0 + ln16;
    bool nv = gn < 656;
    // Clamp so B loads are always in-bounds & unconditional.
    int gnc = nv ? gn : 655;
    int wcol = (gnc < 640) ? (512 + gnc) : (1280 + (gnc - 640));
    const float* wptr = W_in_s + (size_t)kk * 1296 + wcol;
    v8f c = Z8;
#pragma unroll 4
    for (int k0 = 0; k0 < 128; k0 += 4) {
      v2f a, bb;
      a.x = A_s[ml][k0 + kk];
      a.y = A_s[ml][k0 + kk + 1];
      bb.x = wptr[(size_t)k0 * 1296];
      bb.y = wptr[(size_t)k0 * 1296 + 1296];
      c = wmma4(a, bb, c);
    }
#pragma unroll
    for (int e = 0; e < 8; ++e) {
      int gm = m_base + m0 + e + 8 * (lane >> 4);
      if (nv) proj2[((size_t)b * 4096 + gm) * 656 + gn] = c[e];
    }
  }
}

// ===========================================================================
// Kernel 3: stage-2 softplus + scan (8 batches x 16 heads, L=4096 serial).
// ===========================================================================
__global__ void k3_scan(const float* __restrict__ proj2,
                        const float* __restrict__ dt_bias_s,
                        const float* __restrict__ A_log_s,
                        float* __restrict__ w2) {
  const int b = blockIdx.x;
  const int h = threadIdx.x;
  if (h >= 16) return;
  float Ah = -__expf(A_log_s[h]);
  float db = dt_bias_s[h];
  const float* pd = proj2 + (size_t)b * 4096 * 656 + 640 + h;
  float* wb = w2 + (size_t)b * 4096 * 16 + h;
  float Stot = 0.f;
  for (int l = 0; l < 4096; ++l) {
    float dt = softplusf_(pd[(size_t)l * 656] + db);
    wb[(size_t)l * 16] = dt;
    Stot += dt * Ah;
  }
  float S = 0.f;
  for (int l = 0; l < 4096; ++l) {
    float dt = wb[(size_t)l * 16];
    S += dt * Ah;
    wb[(size_t)l * 16] = __expf(Stot - S) * dt;
  }
}

// ===========================================================================
// Kernel 4: stage-2 conv(4)+SiLU, head-reduced xs2 (32 cols) and B2 (128).
// ===========================================================================
__global__ __launch_bounds__(256) void k4_conv(
    const float* __restrict__ proj2, const float* __restrict__ conv_w_s,
    const float* __restrict__ conv_b_s, const float* __restrict__ w2,
    float* __restrict__ xs2, float* __restrict__ B2) {
  long tid = (long)blockIdx.x * 256 + threadIdx.x;
  int q = (int)(tid % 160);
  long bl = tid / 160;          // b*4096 + l
  int l = (int)(bl & 4095);
  const float* pbatch = proj2 + (bl - l) * 656;  // start of this batch
  if (q < 32) {
    float sum = 0.f;
    for (int h = 0; h < 16; ++h) {
      int cg = h * 32 + q;
      float acc = conv_b_s[cg];
#pragma unroll
      for (int k = 0; k < 4; ++k) {
        int ls = l - 3 + k;
        int lsc = (ls >= 0) ? ls : 0;
        float msk = (ls >= 0) ? 1.f : 0.f;
        acc += conv_w_s[cg * 4 + k] * pbatch[(long)lsc * 656 + cg] * msk;
      }
      sum += w2[bl * 16 + h] * siluf_(acc);
    }
    xs2[bl * 32 + q] = sum;
  } else {
    int n = q - 32;
    int cg = 512 + n;
    float acc = conv_b_s[cg];
#pragma unroll
    for (int k = 0; k < 4; ++k) {
      int ls = l - 3 + k;
      int lsc = (ls >= 0) ? ls : 0;
      float msk = (ls >= 0) ? 1.f : 0.f;
      acc += conv_w_s[cg * 4 + k] * pbatch[(long)lsc * 656 + cg] * msk;
    }
    B2[bl * 128 + n] = siluf_(acc);
  }
}

// ===========================================================================
// Kernel 5: feat(32x128) = xs2^T @ B2 per batch (K=4096, 64-chunk LDS stage).
// ===========================================================================
__global__ __launch_bounds__(256) void k5_feat(const float* __restrict__ xs2,
                                               const float* __restrict__ B2,
                                               float* __restrict__ feat) {
  __shared__ float xc_s[64][33];
  __shared__ float Bc_s[64][128];
  const int tid = threadIdx.x, lane = tid & 31, wave = tid >> 5;
  const int b = blockIdx.x;
  const int kk = (lane >> 4) * 2;
  const int ln16 = lane & 15;
  const int nl = wave * 16 + ln16;  // each wave owns one 16-col N tile

  v8f acc0 = Z8, acc1 = Z8;  // M tiles 0 and 1 (p 0..15, 16..31)
  for (int kc = 0; kc < 4096; kc += 64) {
    for (int i = tid; i < 64 * 32; i += 256) {
      int row = i >> 5, col = i & 31;
      xc_s[row][col] = xs2[((size_t)b * 4096 + kc + row) * 32 + col];
    }
    for (int i = tid; i < 64 * 128; i += 256) {
      int row = i >> 7, col = i & 127;
      Bc_s[row][col] = B2[((size_t)b * 4096 + kc + row) * 128 + col];
    }
    __syncthreads();
#pragma unroll 4
    for (int k0 = 0; k0 < 64; k0 += 4) {
      v2f bf, a0, a1;
      bf.x = Bc_s[k0 + kk][nl];
      bf.y = Bc_s[k0 + kk + 1][nl];
      a0.x = xc_s[k0 + kk][ln16];
      a0.y = xc_s[k0 + kk + 1][ln16];
      a1.x = xc_s[k0 + kk][16 + ln16];
      a1.y = xc_s[k0 + kk + 1][16 + ln16];
      acc0 = wmma4(a0, bf, acc0);
      acc1 = wmma4(a1, bf, acc1);
    }
    __syncthreads();
  }
  float* fb = feat + (size_t)b * 4096;
#pragma unroll
  for (int e = 0; e < 8; ++e) {
    int row = e + 8 * (lane >> 4);
    fb[row * 128 + nl] = acc0[e];
    fb[(16 + row) * 128 + nl] = acc1[e];
  }
}

// ===========================================================================
// Kernel 6: classifier y = feat @ W_cls^T + b_cls, broadcast over T.
// ===========================================================================
__global__ __launch_bounds__(64) void k6_cls(const float* __restrict__ feat,
                                             const float* __restrict__ W_cls,
                                             const float* __restrict__ b_cls,
                                             float* __restrict__ out) {
  const int b = blockIdx.x;
  const int c = threadIdx.x;
  if (c >= 60) return;
  const float* fb = feat + (size_t)b * 4096;
  const float* wc = W_cls + (size_t)c * 4096;
  float acc = b_cls[c];
  for (int j = 0; j < 4096; ++j) acc += fb[j] * wc[j];
  for (int t = 0; t < 32; ++t) out[((size_t)b * 32 + t) * 60 + c] = acc;
}

// ===========================================================================
extern "C" void kernel_launch(void* const* d_in, const int* in_sizes, int n_in,
                              void* d_out, int out_size, void* d_ws,
                              size_t ws_size, hipStream_t stream) {
  const float* inputs    = (const float*)d_in[0];
  const float* W_in_r    = (const float*)d_in[1];
  const float* conv_w_r  = (const float*)d_in[2];
  const float* conv_b_r  = (const float*)d_in[3];
  const float* dt_bias_r = (const float*)d_in[4];
  const float* A_log_r   = (const float*)d_in[5];
  const float* W_in_s    = (const float*)d_in[6];
  const float* conv_w_s  = (const float*)d_in[7];
  const float* conv_b_s  = (const float*)d_in[8];
  const float* dt_bias_s = (const float*)d_in[9];
  const float* A_log_s   = (const float*)d_in[10];
  const float* W_cls     = (const float*)d_in[11];
  const float* b_cls     = (const float*)d_in[12];
  float* out = (float*)d_out;

  // Workspace layout (bytes); total ~120.1 MB
  char* ws = (char*)d_ws;
  float* agg   = (float*)(ws);                 // 4*256*32*128   = 16 MB
  float* proj2 = (float*)(ws + 16777216);      // 8*4096*656     = 82 MB
  float* w2    = (float*)(ws + 102760448);     // 8*4096*16      =  2 MB
  float* xs2   = (float*)(ws + 104857600);     // 8*4096*32      =  4 MB
  float* B2    = (float*)(ws + 109051904);     // 8*4096*128     = 16 MB
  float* feat  = (float*)(ws + 125829120);     // 8*4096         = 128 KB

  k1_region<<<1024, 256, 0, stream>>>(inputs, W_in_r, conv_w_r, conv_b_r,
                                      dt_bias_r, A_log_r, agg);
  k2_proj<<<dim3(64, 11, 8), 256, 0, stream>>>(agg, W_in_s, proj2);
  k3_scan<<<8, 16, 0, stream>>>(proj2, dt_bias_s, A_log_s, w2);
  long total4 = (long)8 * 4096 * 160;
  k4_conv<<<dim3((unsigned)((total4 + 255) / 256)), 256, 0, stream>>>(
      proj2, conv_w_s, conv_b_s, w2, xs2, B2);
  k5_feat<<<8, 256, 0, stream>>>(xs2, B2, feat);
  k6_cls<<<8, 64, 0, stream>>>(feat, W_cls, b_cls, out);
}